// ResidualMamba_Block_12884901888673
// MI455X (gfx1250) — compile-verified
//
#include <hip/hip_runtime.h>
#include <hip/hip_bf16.h>

// ---------------------------------------------------------------------------
// Mamba-2 block forward for MI455X (gfx1250, wave32).
// All GEMM-shaped contractions go through V_WMMA_F32_16X16X4_F32 (fp32 in/out
// keeps reference numerics).  ~121 GFLOP on ~0.5GB -> compute bound, so the
// matrix pipes are the target; tile staging uses CDNA5 async-to-LDS copies
// (ASYNCcnt) double-buffered against the WMMA stream.
// ---------------------------------------------------------------------------

typedef __attribute__((ext_vector_type(2))) float v2f;
typedef __attribute__((ext_vector_type(8))) float v8f;

#define BATCH   4
#define SEQ     2048
#define DMODEL  1024
#define DINNER  2048
#define NHEADS  32
#define HEADDIM 64
#define DSTATE  128
#define CONVCH  2304        // DINNER + 2*DSTATE
#define DINPROJ 4384        // 2*DINNER + 2*DSTATE + NHEADS
#define CHUNK   128
#define NCHUNK  16          // SEQ / CHUNK
#define ROWS    (BATCH*SEQ) // 8192
#define LDP     36          // LDS row pitch (floats): 144B -> 16B aligned rows,
                            // conflict-free for 16-lane stride-LDP reads

__device__ inline v8f wmma4(v2f a, v2f b, v8f c) {
  // D = A(16x4) * B(4x16) + C, fp32 throughout.
  return __builtin_amdgcn_wmma_f32_16x16x4_f32(false, a, false, b,
                                               (short)0, c, false, false);
}

typedef __attribute__((address_space(3))) void lds_void;
__device__ inline unsigned lds_off(void* p) {
  // generic -> LDS addrspace cast yields the 32-bit LDS byte offset
  return (unsigned)(unsigned long long)(lds_void*)p;
}

__device__ inline void async_copy16(void* lds_dst, const void* gsrc) {
  unsigned d = lds_off(lds_dst);
  unsigned long long s = (unsigned long long)(uintptr_t)gsrc;
  asm volatile("global_load_async_to_lds_b128 %0, %1, off"
               :: "v"(d), "v"(s) : "memory");
}

__device__ inline void async_wait0() {
  asm volatile("s_wait_asynccnt 0x0" ::: "memory");
}

__device__ inline float block_reduce_sum(float v, float* red) {
  int tid = threadIdx.x;
  red[tid] = v; __syncthreads();
  for (int s = 128; s > 0; s >>= 1) {
    if (tid < s) red[tid] += red[tid + s];
    __syncthreads();
  }
  float r = red[0]; __syncthreads();
  return r;
}

// ---------------------------------------------------------------------------
// 1) LayerNorm: h = (x - mu)/sqrt(var+eps)*w + b      (one block per row)
// ---------------------------------------------------------------------------
__global__ __launch_bounds__(256)
void ln_kernel(const float* __restrict__ x, const float* __restrict__ w,
               const float* __restrict__ b, float* __restrict__ h) {
  __shared__ float red[256];
  int row = blockIdx.x, tid = threadIdx.x;
  const float* xr = x + (size_t)row * DMODEL;
  float v[4], s = 0.f;
#pragma unroll
  for (int i = 0; i < 4; ++i) { v[i] = xr[tid + i * 256]; s += v[i]; }
  float mu = block_reduce_sum(s, red) * (1.f / DMODEL);
  s = 0.f;
#pragma unroll
  for (int i = 0; i < 4; ++i) { float d = v[i] - mu; s += d * d; }
  float var = block_reduce_sum(s, red) * (1.f / DMODEL);
  float rs = rsqrtf(var + 1e-5f);
  float* hr = h + (size_t)row * DMODEL;
#pragma unroll
  for (int i = 0; i < 4; ++i) {
    int j = tid + i * 256;
    hr[j] = (v[i] - mu) * rs * w[j] + b[j];
  }
}

// ---------------------------------------------------------------------------
// 2) Generic batched WMMA GEMM:  C[m,n] = sum_k A[m,k] * W[n,k]  (+ res)
//    Block tile 64x32, BK=32; 8 waves (4x2) x one 16x16 WMMA tile each.
//    Tiles staged with GLOBAL_LOAD_ASYNC_TO_LDS_B128, double-buffered.
// ---------------------------------------------------------------------------
__global__ __launch_bounds__(256)
void gemm_nt_kernel(const float* __restrict__ A, const float* __restrict__ W,
                    float* __restrict__ C, const float* __restrict__ res,
                    int M, int N, int K, int lda, int ldw, int ldc,
                    long long sA, long long sW, long long sC) {
  __shared__ __align__(16) float As[2][64][LDP];   // [buf][m][k]
  __shared__ __align__(16) float Bs[2][32][LDP];   // [buf][n][k]
  A += (long long)blockIdx.z * sA;
  W += (long long)blockIdx.z * sW;
  C += (long long)blockIdx.z * sC;
  if (res) res += (long long)blockIdx.z * sC;
  int bm = blockIdx.x * 64, bn = blockIdx.y * 32;
  int tid = threadIdx.x, wave = tid >> 5, lane = tid & 31;
  int wr = wave >> 1, wc = wave & 1;
  int lm = lane & 15, half = (lane >> 4) << 1, up8 = (lane >> 4) << 3;

  auto stage = [&](int buf, int kb) {
    // A tile: 64 rows x 32 floats = 512 b128 transfers, 2 per thread.
#pragma unroll
    for (int j = 0; j < 2; ++j) {
      int idx = tid + j * 256;
      int r = idx >> 3, c4 = (idx & 7) << 2;
      async_copy16(&As[buf][r][c4], &A[(size_t)(bm + r) * lda + kb + c4]);
    }
    // B tile: 32 rows x 32 floats = 256 b128 transfers, 1 per thread.
    {
      int r = tid >> 3, c4 = (tid & 7) << 2;
      async_copy16(&Bs[buf][r][c4], &W[(size_t)(bn + r) * ldw + kb + c4]);
    }
  };

  v8f acc = {0.f,0.f,0.f,0.f,0.f,0.f,0.f,0.f};
  int nk = K >> 5;
  stage(0, 0);
  async_wait0();
  __syncthreads();
  for (int i = 0; i < nk; ++i) {
    int buf = i & 1;
    if (i + 1 < nk) stage(buf ^ 1, (i + 1) << 5);   // overlap with WMMA
#pragma unroll
    for (int k0 = 0; k0 < 32; k0 += 4) {
      v2f a, b;
      a.x = As[buf][wr * 16 + lm][k0 + half];
      a.y = As[buf][wr * 16 + lm][k0 + half + 1];
      b.x = Bs[buf][wc * 16 + lm][k0 + half];
      b.y = Bs[buf][wc * 16 + lm][k0 + half + 1];
      acc = wmma4(a, b, acc);
    }
    async_wait0();      // next tile landed in LDS
    __syncthreads();    // all waves done reading buf / writing buf^1
  }
  int rbase = bm + wr * 16 + up8;
  int cidx = bn + wc * 16 + lm;
#pragma unroll
  for (int r = 0; r < 8; ++r) {
    size_t off = (size_t)(rbase + r) * ldc + cidx;
    float v = acc[r];
    if (res) v += res[off];
    C[off] = v;
  }
}

// ---------------------------------------------------------------------------
// 3a) dt = softplus(zxbcdt[..., -NHEADS:] + dt_bias)
// ---------------------------------------------------------------------------
__global__ __launch_bounds__(256)
void dt_kernel(const float* __restrict__ zx, const float* __restrict__ dtbias,
               float* __restrict__ dtb) {
  int idx = blockIdx.x * 256 + threadIdx.x;       // ROWS*NHEADS
  int row = idx >> 5, h = idx & 31;
  float v = zx[(size_t)row * DINPROJ + (DINNER + CONVCH) + h] + dtbias[h];
  dtb[idx] = (v > 20.f) ? v : log1pf(expf(v));
}

// 3b) per-chunk inclusive cumsum of a = dt * A, and chunk totals dA.
__global__ __launch_bounds__(64)
void acs_kernel(const float* __restrict__ dtb, const float* __restrict__ A_log,
                float* __restrict__ acsb, float* __restrict__ dAb) {
  int t = blockIdx.x * 64 + threadIdx.x;          // BATCH*NCHUNK*NHEADS = 2048
  int h = t & 31, bc = t >> 5;
  size_t g0 = (size_t)bc * CHUNK;
  float negA = -expf(A_log[h]);
  float run = 0.f;
  for (int l = 0; l < CHUNK; ++l) {
    run += dtb[(g0 + l) * NHEADS + h] * negA;
    acsb[(g0 + l) * NHEADS + h] = run;
  }
  dAb[t] = run;
}

// 3c) depthwise conv (k=4, causal) + SiLU on the xBC slice of zxbcdt.
__global__ __launch_bounds__(256)
void conv_kernel(const float* __restrict__ zx, const float* __restrict__ cw,
                 const float* __restrict__ cb, float* __restrict__ xbc) {
  size_t idx = (size_t)blockIdx.x * 256 + threadIdx.x;   // ROWS*CONVCH
  int ch = (int)(idx % CONVCH);
  int row = (int)(idx / CONVCH);
  int b = row >> 11, l = row & 2047;
  float acc = cb[ch];
#pragma unroll
  for (int k = 0; k < 4; ++k) {
    int lp = l + k - 3;
    if (lp >= 0)
      acc += zx[(size_t)((b << 11) + lp) * DINPROJ + DINNER + ch] * cw[ch * 4 + k];
  }
  xbc[(size_t)row * CONVCH + ch] = acc / (1.f + expf(-acc));
}

// ---------------------------------------------------------------------------
// 4b) Per (b,chunk,head): Y_diag = (C·Bᵀ ∘ L) @ (x*dt)  and
//     states = Bᵀ @ (exp(dA-acs) ∘ x*dt).   Decay/mask folded into A-frags;
//     X*dt staged in LDS is the shared B operand of both WMMA chains.
// ---------------------------------------------------------------------------
__global__ __launch_bounds__(256)
void ssd_diag_kernel(const float* __restrict__ xbc, const float* __restrict__ dtb,
                     const float* __restrict__ acsb, const float* __restrict__ dAb,
                     const float* __restrict__ CB, const float* __restrict__ Dp,
                     float* __restrict__ Y, float* __restrict__ states) {
  __shared__ float Xs[CHUNK][HEADDIM];
  __shared__ float acs_s[CHUNK];
  __shared__ float dec_s[CHUNK];
  int idx = blockIdx.x;                 // bc*32 + h
  int h = idx & 31, bc = idx >> 5;
  size_t g0 = (size_t)bc * CHUNK;
  int tid = threadIdx.x;
  float dAv = dAb[idx];
  for (int i = tid; i < CHUNK * HEADDIM; i += 256) {
    int l = i >> 6, p = i & 63;
    Xs[l][p] = xbc[(g0 + l) * CONVCH + h * HEADDIM + p] *
               dtb[(g0 + l) * NHEADS + h];
  }
  for (int i = tid; i < CHUNK; i += 256) {
    float v = acsb[(g0 + i) * NHEADS + h];
    acs_s[i] = v;
    dec_s[i] = expf(dAv - v);
  }
  __syncthreads();
  int wave = tid >> 5, lane = tid & 31;
  int lm = lane & 15, half = (lane >> 4) << 1, up8 = (lane >> 4) << 3;
  int lrow = wave * 16 + lm;            // l for Y_diag rows, n for state rows
  float acs_l = acs_s[lrow];
  const float* CBp = CB + (size_t)bc * CHUNK * CHUNK;
  const float* Bm = xbc + g0 * CONVCH + DINNER;     // Bm[l][n]
  v8f z8 = {0.f,0.f,0.f,0.f,0.f,0.f,0.f,0.f};
  v8f aY[4] = {z8, z8, z8, z8};
  v8f aS[4] = {z8, z8, z8, z8};
  for (int k0 = 0; k0 < CHUNK; k0 += 4) {
    int s0 = k0 + half;
    v2f a, as;
    a.x = (s0     <= lrow) ? CBp[lrow * CHUNK + s0    ] * expf(acs_l - acs_s[s0    ]) : 0.f;
    a.y = (s0 + 1 <= lrow) ? CBp[lrow * CHUNK + s0 + 1] * expf(acs_l - acs_s[s0 + 1]) : 0.f;
    as.x = Bm[(size_t)s0 * CONVCH + lrow] * dec_s[s0];
    as.y = Bm[(size_t)(s0 + 1) * CONVCH + lrow] * dec_s[s0 + 1];
#pragma unroll
    for (int ct = 0; ct < 4; ++ct) {
      v2f bf;
      bf.x = Xs[s0][ct * 16 + lm];
      bf.y = Xs[s0 + 1][ct * 16 + lm];
      aY[ct] = wmma4(a, bf, aY[ct]);
      aS[ct] = wmma4(as, bf, aS[ct]);
    }
  }
  float Dh = Dp[h];
  float* Sp = states + (size_t)idx * (DSTATE * HEADDIM);
#pragma unroll
  for (int ct = 0; ct < 4; ++ct) {
#pragma unroll
    for (int r = 0; r < 8; ++r) {
      int row = wave * 16 + up8 + r;
      int col = ct * 16 + lm;
      size_t yoff = (g0 + row) * DINNER + h * HEADDIM + col;
      Y[yoff] = aY[ct][r] +
                Dh * xbc[(g0 + row) * CONVCH + h * HEADDIM + col];
      Sp[(size_t)row * HEADDIM + col] = aS[ct][r];
    }
  }
}

// ---------------------------------------------------------------------------
// 5) Inter-chunk scan (per b,h): prev[c]=P; P = exp(dA[c])*P + states[c]
// ---------------------------------------------------------------------------
__global__ __launch_bounds__(256)
void scan_kernel(const float* __restrict__ states, const float* __restrict__ dAb,
                 float* __restrict__ prev) {
  int bh = blockIdx.x;                  // b*32 + h
  int b = bh >> 5, h = bh & 31;
  int tid = threadIdx.x;
  float P[32];
#pragma unroll
  for (int i = 0; i < 32; ++i) P[i] = 0.f;
  for (int c = 0; c < NCHUNK; ++c) {
    int sidx = (b * NCHUNK + c) * NHEADS + h;
    float e = expf(dAb[sidx]);
    const float* st = states + (size_t)sidx * (DSTATE * HEADDIM);
    float* pv = prev + (size_t)sidx * (DSTATE * HEADDIM);
#pragma unroll
    for (int i = 0; i < 32; ++i) {
      int e2 = i * 256 + tid;
      pv[e2] = P[i];
      P[i] = e * P[i] + st[e2];
    }
  }
}

// ---------------------------------------------------------------------------
// 6) Y += exp(acs[l]) * (C @ prev_state)    (WMMA, exp folded into A-frag)
// ---------------------------------------------------------------------------
__global__ __launch_bounds__(256)
void ssd_off_kernel(const float* __restrict__ xbc, const float* __restrict__ acsb,
                    const float* __restrict__ prev, float* __restrict__ Y) {
  __shared__ float acs_s[CHUNK];
  int idx = blockIdx.x;                 // bc*32 + h
  int h = idx & 31, bc = idx >> 5;
  size_t g0 = (size_t)bc * CHUNK;
  int tid = threadIdx.x;
  for (int i = tid; i < CHUNK; i += 256)
    acs_s[i] = acsb[(g0 + i) * NHEADS + h];
  __syncthreads();
  int wave = tid >> 5, lane = tid & 31;
  int lm = lane & 15, half = (lane >> 4) << 1, up8 = (lane >> 4) << 3;
  int lrow = wave * 16 + lm;
  float e_l = expf(acs_s[lrow]);
  const float* Cm = xbc + g0 * CONVCH + DINNER + DSTATE;   // Cm[l][n]
  const float* Pv = prev + (size_t)idx * (DSTATE * HEADDIM);
  v8f z8 = {0.f,0.f,0.f,0.f,0.f,0.f,0.f,0.f};
  v8f acc[4] = {z8, z8, z8, z8};
  for (int k0 = 0; k0 < DSTATE; k0 += 4) {
    int s0 = k0 + half;
    v2f a;
    a.x = Cm[(size_t)lrow * CONVCH + s0] * e_l;
    a.y = Cm[(size_t)lrow * CONVCH + s0 + 1] * e_l;
#pragma unroll
    for (int ct = 0; ct < 4; ++ct) {
      v2f bf;
      bf.x = Pv[(size_t)s0 * HEADDIM + ct * 16 + lm];
      bf.y = Pv[(size_t)(s0 + 1) * HEADDIM + ct * 16 + lm];
      acc[ct] = wmma4(a, bf, acc[ct]);
    }
  }
#pragma unroll
  for (int ct = 0; ct < 4; ++ct) {
#pragma unroll
    for (int r = 0; r < 8; ++r) {
      int row = wave * 16 + up8 + r;
      int col = ct * 16 + lm;
      Y[(g0 + row) * DINNER + h * HEADDIM + col] += acc[ct][r];
    }
  }
}

// ---------------------------------------------------------------------------
// 7) y = Y * silu(z); y *= rsqrt(mean(y^2)+eps) * norm_w   (in place on Y)
// ---------------------------------------------------------------------------
__global__ __launch_bounds__(256)
void gate_rms_kernel(const float* __restrict__ zx, const float* __restrict__ nw,
                     float* __restrict__ Y) {
  __shared__ float red[256];
  int row = blockIdx.x, tid = threadIdx.x;
  float v[8], ss = 0.f;
#pragma unroll
  for (int i = 0; i < 8; ++i) {
    int j = tid + i * 256;
    float z = zx[(size_t)row * DINPROJ + j];
    float val = Y[(size_t)row * DINNER + j] * (z / (1.f + expf(-z)));
    v[i] = val;
    ss += val * val;
  }
  float mean = block_reduce_sum(ss, red) * (1.f / DINNER);
  float sc = rsqrtf(mean + 1e-5f);
#pragma unroll
  for (int i = 0; i < 8; ++i) {
    int j = tid + i * 256;
    Y[(size_t)row * DINNER + j] = v[i] * sc * nw[j];
  }
}

// ---------------------------------------------------------------------------
// Launch
// ---------------------------------------------------------------------------
extern "C" void kernel_launch(void* const* d_in, const int* in_sizes, int n_in,
                              void* d_out, int out_size, void* d_ws, size_t ws_size,
                              hipStream_t stream) {
  const float* x      = (const float*)d_in[0];
  const float* ln_w   = (const float*)d_in[1];
  const float* ln_b   = (const float*)d_in[2];
  const float* w_in   = (const float*)d_in[3];   // [4384,1024]
  const float* conv_w = (const float*)d_in[4];   // [2304,4]
  const float* conv_b = (const float*)d_in[5];
  const float* dtbias = (const float*)d_in[6];
  const float* A_log  = (const float*)d_in[7];
  const float* Dp     = (const float*)d_in[8];
  const float* norm_w = (const float*)d_in[9];
  const float* w_out  = (const float*)d_in[10];  // [1024,2048]
  float* out = (float*)d_out;
  float* ws  = (float*)d_ws;

  size_t o = 0;
  float* h    = ws + o; o += (size_t)ROWS * DMODEL;            // 8192x1024
  float* zx   = ws + o; o += (size_t)ROWS * DINPROJ;           // 8192x4384
  float* xbc  = ws + o; o += (size_t)ROWS * CONVCH;            // 8192x2304
  float* dtb  = ws + o; o += (size_t)ROWS * NHEADS;
  float* acsb = ws + o; o += (size_t)ROWS * NHEADS;
  float* dAb  = ws + o; o += (size_t)BATCH * NCHUNK * NHEADS;
  float* CBb  = ws + o; o += (size_t)BATCH * NCHUNK * CHUNK * CHUNK;
  float* Yb   = ws + o; o += (size_t)ROWS * DINNER;
  float* stb  = ws + o; o += (size_t)BATCH * NCHUNK * NHEADS * DSTATE * HEADDIM;
  float* pvb  = ws + o; o += (size_t)BATCH * NCHUNK * NHEADS * DSTATE * HEADDIM;
  // total ~115M floats (~460MB) required in d_ws

  // 1) LayerNorm
  ln_kernel<<<ROWS, 256, 0, stream>>>(x, ln_w, ln_b, h);

  // 2) in_proj: zx = h @ w_in^T     (WMMA f32, async-LDS double buffered)
  gemm_nt_kernel<<<dim3(ROWS / 64, DINPROJ / 32, 1), 256, 0, stream>>>(
      h, w_in, zx, nullptr, ROWS, DINPROJ, DMODEL, DMODEL, DMODEL, DINPROJ,
      0, 0, 0);

  // 3) dt / cumsum / conv+silu
  dt_kernel<<<ROWS * NHEADS / 256, 256, 0, stream>>>(zx, dtbias, dtb);
  acs_kernel<<<BATCH * NCHUNK * NHEADS / 64, 64, 0, stream>>>(dtb, A_log, acsb, dAb);
  conv_kernel<<<(unsigned)(((size_t)ROWS * CONVCH) / 256), 256, 0, stream>>>(
      zx, conv_w, conv_b, xbc);

  // 4a) per-chunk CB = C @ B^T  (batched WMMA GEMM, batch = b*c)
  gemm_nt_kernel<<<dim3(CHUNK / 64, CHUNK / 32, BATCH * NCHUNK), 256, 0, stream>>>(
      xbc + DINNER + DSTATE, xbc + DINNER, CBb, nullptr,
      CHUNK, DSTATE, DSTATE, CONVCH, CONVCH, CHUNK,
      (long long)CHUNK * CONVCH, (long long)CHUNK * CONVCH,
      (long long)CHUNK * CHUNK);

  // 4b) intra-chunk Y_diag + chunk states (WMMA)
  ssd_diag_kernel<<<BATCH * NCHUNK * NHEADS, 256, 0, stream>>>(
      xbc, dtb, acsb, dAb, CBb, Dp, Yb, stb);

  // 5) inter-chunk scan
  scan_kernel<<<BATCH * NHEADS, 256, 0, stream>>>(stb, dAb, pvb);

  // 6) off-diagonal contribution (WMMA)
  ssd_off_kernel<<<BATCH * NCHUNK * NHEADS, 256, 0, stream>>>(xbc, acsb, pvb, Yb);

  // 7) gate + RMSNorm (in place on Yb)
  gate_rms_kernel<<<ROWS, 256, 0, stream>>>(zx, norm_w, Yb);

  // 8) out_proj + residual: out = Yb @ w_out^T + x   (WMMA f32)
  gemm_nt_kernel<<<dim3(ROWS / 64, DMODEL / 32, 1), 256, 0, stream>>>(
      Yb, w_out, out, x, ROWS, DMODEL, DINNER, DINNER, DINNER, DMODEL,
      0, 0, 0);
}